// PerHeadConvNetLayer_78580721647666
// MI455X (gfx1250) — compile-verified
//
#include <hip/hip_runtime.h>
#include <math.h>

typedef __attribute__((ext_vector_type(2))) float v2f;
typedef __attribute__((ext_vector_type(8))) float v8f;

#define NNODES 20000
#define NEDGES 640000

// D = A(16x4) * B(4x16) + C, all fp32
__device__ __forceinline__ v8f wmma4(v2f a, v2f b, v8f c) {
  return __builtin_amdgcn_wmma_f32_16x16x4_f32(false, a, false, b, (short)0, c,
                                               false, false);
}

// B fragment: W is row-major KxN (ld = row stride). 4x16 tile at (k0, n0).
// lanes 0-15: K=k0,k0+1 ; lanes 16-31: K=k0+2,k0+3 ; N = n0 + (lane&15)
__device__ __forceinline__ v2f load_b4(const float* W, int ld, int k0, int n0,
                                       int lane) {
  int n = n0 + (lane & 15);
  int k = k0 + ((lane >> 4) << 1);
  v2f b;
  b.x = W[k * ld + n];
  b.y = W[(k + 1) * ld + n];
  return b;
}

// silu via v_rcp_f32 (one TRANS op) instead of IEEE division chain
__device__ __forceinline__ float silu(float v) {
  return v * __builtin_amdgcn_rcpf(1.0f + __expf(-v));
}

__global__ void k_zero(float* __restrict__ p, int n) {
  int i = blockIdx.x * blockDim.x + threadIdx.x;
  int stride = gridDim.x * blockDim.x;
  for (; i < n; i += stride) p[i] = 0.0f;
}

// ---------------- Node precompute: y0,y1,y2 and sc (one wave per node) -----
__global__ void __launch_bounds__(256)
k_node_pre(const float* __restrict__ x, const float* __restrict__ na,
           const float* __restrict__ W1_0, const float* __restrict__ W1_1,
           const float* __restrict__ W1_2, const float* __restrict__ scW,
           float* __restrict__ y0ws, float* __restrict__ y1ws,
           float* __restrict__ y2ws, float* __restrict__ scws) {
  int lane = threadIdx.x & 31;
  int node = blockIdx.x * (blockDim.x >> 5) + (threadIdx.x >> 5);
  if (node >= NNODES) return;
  const float inv_mul = 0.17677669529663687f;  // 1/sqrt(32)
  const float* xr = x + node * 288;

  float acc = 0.f;
#pragma unroll 8
  for (int k = 0; k < 32; ++k) acc = fmaf(xr[k], W1_0[k * 32 + lane], acc);
  y0ws[node * 32 + lane] = acc * inv_mul;

#pragma unroll
  for (int m = 0; m < 3; ++m) {
    acc = 0.f;
    for (int k = 0; k < 32; ++k)
      acc = fmaf(xr[32 + k * 3 + m], W1_1[k * 32 + lane], acc);
    y1ws[node * 96 + lane * 3 + m] = acc * inv_mul;
  }
#pragma unroll
  for (int m = 0; m < 5; ++m) {
    acc = 0.f;
    for (int k = 0; k < 32; ++k)
      acc = fmaf(xr[128 + k * 5 + m], W1_2[k * 32 + lane], acc);
    y2ws[node * 160 + lane * 5 + m] = acc * inv_mul;
  }
  // sc[h,n,w] = sum_{u,v} x0[n,u]*na[n,v]*scW[h,u,v,w] / sqrt(128)
  float nav0 = na[node * 4 + 0], nav1 = na[node * 4 + 1];
  float nav2 = na[node * 4 + 2], nav3 = na[node * 4 + 3];
  const float inv_sc = 0.08838834764831845f;  // 1/sqrt(128)
  for (int h = 0; h < 3; ++h) {
    acc = 0.f;
    for (int u = 0; u < 32; ++u) {
      const float* wp = scW + ((h * 32 + u) * 4) * 32 + lane;
      float t = nav0 * wp[0] + nav1 * wp[32] + nav2 * wp[64] + nav3 * wp[96];
      acc = fmaf(xr[u], t, acc);
    }
    scws[(h * NNODES + node) * 32 + lane] = acc * inv_sc;
  }
}

// ---------------- Edge MLP (WMMA fp32) + messages + scatter ----------------
// One wave handles 16 edges. 4 waves / block.
__global__ void __launch_bounds__(128)
k_edge(const float* __restrict__ eemb, const float* __restrict__ eattr,
       const int* __restrict__ eidx, const float* __restrict__ Wm0,
       const float* __restrict__ Wm1, const float* __restrict__ y0ws,
       const float* __restrict__ y1ws, const float* __restrict__ y2ws,
       float* __restrict__ mid) {
  __shared__ float lds[4 * (16 * 65 + 16 * 97)];
  int lane = threadIdx.x & 31;
  int wv = threadIdx.x >> 5;
  int tile = blockIdx.x * 4 + wv;
  int e0 = tile * 16;
  float* hbuf = lds + wv * (16 * 65 + 16 * 97);
  float* wbuf = hbuf + 16 * 65;
  int half = lane >> 4;
  int mrow = lane & 15;

  // speculative prefetch of the next tile's edge data (global_prefetch_b8)
  if (e0 + 64 < NEDGES) {
    __builtin_prefetch(eemb + (e0 + 64) * 8, 0, 1);
    __builtin_prefetch(eattr + (e0 + 64) * 9, 0, 1);
  }

  // ---- layer 1: (16x8) @ (8x64), K split into two x4 steps ----
  v8f a0 = {}, a1 = {}, a2 = {}, a3 = {};
#pragma unroll
  for (int kk = 0; kk < 8; kk += 4) {
    v2f a;
    const float* ap = eemb + (e0 + mrow) * 8 + kk + half * 2;
    a.x = ap[0];
    a.y = ap[1];
    a0 = wmma4(a, load_b4(Wm0, 64, kk, 0, lane), a0);
    a1 = wmma4(a, load_b4(Wm0, 64, kk, 16, lane), a1);
    a2 = wmma4(a, load_b4(Wm0, 64, kk, 32, lane), a2);
    a3 = wmma4(a, load_b4(Wm0, 64, kk, 48, lane), a3);
  }
  const float s1 = 0.35355339059327373f;  // 1/sqrt(8)
#pragma unroll
  for (int r = 0; r < 8; ++r) {
    int row = r + 8 * half;
    hbuf[row * 65 + mrow + 0]  = silu(a0[r] * s1);
    hbuf[row * 65 + mrow + 16] = silu(a1[r] * s1);
    hbuf[row * 65 + mrow + 32] = silu(a2[r] * s1);
    hbuf[row * 65 + mrow + 48] = silu(a3[r] * s1);
  }
  __syncthreads();

  // ---- layer 2: (16x64) @ (64x96) ----
  v8f w0 = {}, w1 = {}, w2 = {}, w3 = {}, w4 = {}, w5 = {};
#pragma unroll 4
  for (int kk = 0; kk < 64; kk += 4) {
    v2f a;
    int k = kk + half * 2;
    a.x = hbuf[mrow * 65 + k];
    a.y = hbuf[mrow * 65 + k + 1];
    w0 = wmma4(a, load_b4(Wm1, 96, kk, 0, lane), w0);
    w1 = wmma4(a, load_b4(Wm1, 96, kk, 16, lane), w1);
    w2 = wmma4(a, load_b4(Wm1, 96, kk, 32, lane), w2);
    w3 = wmma4(a, load_b4(Wm1, 96, kk, 48, lane), w3);
    w4 = wmma4(a, load_b4(Wm1, 96, kk, 64, lane), w4);
    w5 = wmma4(a, load_b4(Wm1, 96, kk, 80, lane), w5);
  }
  const float s2 = 0.125f;  // 1/sqrt(64)
#pragma unroll
  for (int r = 0; r < 8; ++r) {
    int row = r + 8 * half;
    wbuf[row * 97 + mrow + 0]  = w0[r] * s2;
    wbuf[row * 97 + mrow + 16] = w1[r] * s2;
    wbuf[row * 97 + mrow + 32] = w2[r] * s2;
    wbuf[row * 97 + mrow + 48] = w3[r] * s2;
    wbuf[row * 97 + mrow + 64] = w4[r] * s2;
    wbuf[row * 97 + mrow + 80] = w5[r] * s2;
  }
  __syncthreads();

  // ---- messages + scatter-add (lane = u channel) ----
  const float inv3 = 0.5773502691896258f;      // 1/sqrt(3)
  const float inv5 = 0.4472135954999579f;      // 1/sqrt(5)
  const float inv_avg = 0.17677669529663687f;  // 1/sqrt(32)
  for (int e = 0; e < 16; ++e) {
    int ge = e0 + e;
    int sn = eidx[ge];
    int dn = eidx[NEDGES + ge];
    const float* ea = eattr + ge * 9;
    float m0v = wbuf[e * 97 + lane] * y0ws[sn * 32 + lane] * ea[0];
    const float* y1p = y1ws + sn * 96 + lane * 3;
    float d1 = y1p[0] * ea[1] + y1p[1] * ea[2] + y1p[2] * ea[3];
    float m1v = wbuf[e * 97 + 32 + lane] * d1 * inv3;
    const float* y2p = y2ws + sn * 160 + lane * 5;
    float d2 = y2p[0] * ea[4] + y2p[1] * ea[5] + y2p[2] * ea[6] +
               y2p[3] * ea[7] + y2p[4] * ea[8];
    float m2v = wbuf[e * 97 + 64 + lane] * d2 * inv5;
    float* mp = mid + (long)dn * 96;
    unsafeAtomicAdd(mp + lane, m0v * inv_avg);
    unsafeAtomicAdd(mp + 32 + lane, m1v * inv_avg);
    unsafeAtomicAdd(mp + 64 + lane, m2v * inv_avg);
  }
}

// ---------------- Final node update (WMMA fp32) ----------------------------
__global__ void __launch_bounds__(128)
k_node_out(const float* __restrict__ mid, const float* __restrict__ l20,
           const float* __restrict__ l21, const float* __restrict__ l22,
           const float* __restrict__ scws, float* __restrict__ out) {
  int lane = threadIdx.x & 31;
  int wv = threadIdx.x >> 5;
  int tile = blockIdx.x * 4 + wv;
  if (tile * 16 >= NNODES) return;  // wave-uniform exit
  int n0 = tile * 16;
  int half = lane >> 4;
  int mrow = lane & 15;

  v8f c00 = {}, c01 = {}, c10 = {}, c11 = {}, c20 = {}, c21 = {};
  // Phase 1: K = 0..32 feeds all three heads
#pragma unroll
  for (int kk = 0; kk < 32; kk += 4) {
    v2f a;
    int k = kk + half * 2;
    const float* ap = mid + (long)(n0 + mrow) * 96 + k;
    a.x = ap[0];
    a.y = ap[1];
    c00 = wmma4(a, load_b4(l20, 32, kk, 0, lane), c00);
    c01 = wmma4(a, load_b4(l20, 32, kk, 16, lane), c01);
    c10 = wmma4(a, load_b4(l21, 32, kk, 0, lane), c10);
    c11 = wmma4(a, load_b4(l21, 32, kk, 16, lane), c11);
    c20 = wmma4(a, load_b4(l22, 32, kk, 0, lane), c20);
    c21 = wmma4(a, load_b4(l22, 32, kk, 16, lane), c21);
  }
  // Phase 2: K = 32..64 feeds heads 1 and 2
#pragma unroll
  for (int kk = 32; kk < 64; kk += 4) {
    v2f a;
    int k = kk + half * 2;
    const float* ap = mid + (long)(n0 + mrow) * 96 + k;
    a.x = ap[0];
    a.y = ap[1];
    c10 = wmma4(a, load_b4(l21, 32, kk, 0, lane), c10);
    c11 = wmma4(a, load_b4(l21, 32, kk, 16, lane), c11);
    c20 = wmma4(a, load_b4(l22, 32, kk, 0, lane), c20);
    c21 = wmma4(a, load_b4(l22, 32, kk, 16, lane), c21);
  }
  // Phase 3: K = 64..96 feeds head 2 only
#pragma unroll
  for (int kk = 64; kk < 96; kk += 4) {
    v2f a;
    int k = kk + half * 2;
    const float* ap = mid + (long)(n0 + mrow) * 96 + k;
    a.x = ap[0];
    a.y = ap[1];
    c20 = wmma4(a, load_b4(l22, 32, kk, 0, lane), c20);
    c21 = wmma4(a, load_b4(l22, 32, kk, 16, lane), c21);
  }
  const float s0 = 0.17677669529663687f;  // 1/sqrt(32)
  const float s1 = 0.125f;                // 1/sqrt(64)
  const float s2 = 0.10206207261596575f;  // 1/sqrt(96)
#pragma unroll
  for (int r = 0; r < 8; ++r) {
    int row = n0 + r + 8 * half;
    int clo = mrow, chi = 16 + mrow;
    out[0 * NNODES * 32 + row * 32 + clo] =
        silu(c00[r] * s0 + scws[(0 * NNODES + row) * 32 + clo]);
    out[0 * NNODES * 32 + row * 32 + chi] =
        silu(c01[r] * s0 + scws[(0 * NNODES + row) * 32 + chi]);
    out[1 * NNODES * 32 + row * 32 + clo] =
        silu(c10[r] * s1 + scws[(1 * NNODES + row) * 32 + clo]);
    out[1 * NNODES * 32 + row * 32 + chi] =
        silu(c11[r] * s1 + scws[(1 * NNODES + row) * 32 + chi]);
    out[2 * NNODES * 32 + row * 32 + clo] =
        silu(c20[r] * s2 + scws[(2 * NNODES + row) * 32 + clo]);
    out[2 * NNODES * 32 + row * 32 + chi] =
        silu(c21[r] * s2 + scws[(2 * NNODES + row) * 32 + chi]);
  }
}

extern "C" void kernel_launch(void* const* d_in, const int* in_sizes, int n_in,
                              void* d_out, int out_size, void* d_ws,
                              size_t ws_size, hipStream_t stream) {
  const float* x = (const float*)d_in[0];
  const float* na = (const float*)d_in[1];
  const float* eemb = (const float*)d_in[2];
  const float* eattr = (const float*)d_in[3];
  const int* eidx = (const int*)d_in[4];
  const float* W1_0 = (const float*)d_in[5];
  const float* W1_1 = (const float*)d_in[6];
  const float* W1_2 = (const float*)d_in[7];
  const float* Wm0 = (const float*)d_in[8];
  const float* Wm1 = (const float*)d_in[9];
  const float* l20 = (const float*)d_in[10];
  const float* l21 = (const float*)d_in[11];
  const float* l22 = (const float*)d_in[12];
  const float* scW = (const float*)d_in[13];

  float* ws = (float*)d_ws;
  float* y0ws = ws;                     // N*32
  float* y1ws = y0ws + NNODES * 32;     // N*96
  float* y2ws = y1ws + NNODES * 96;     // N*160
  float* scws = y2ws + NNODES * 160;    // 3*N*32
  float* mid = scws + 3 * NNODES * 32;  // N*96
  float* out = (float*)d_out;

  k_zero<<<2048, 256, 0, stream>>>(mid, NNODES * 96);
  k_node_pre<<<(NNODES + 7) / 8, 256, 0, stream>>>(x, na, W1_0, W1_1, W1_2,
                                                   scW, y0ws, y1ws, y2ws, scws);
  k_edge<<<NEDGES / 16 / 4, 128, 0, stream>>>(eemb, eattr, eidx, Wm0, Wm1,
                                              y0ws, y1ws, y2ws, mid);
  k_node_out<<<(NNODES / 16 + 3) / 4, 128, 0, stream>>>(mid, l20, l21, l22,
                                                        scws, out);
}